// MLP_learner_9809705304349
// MI455X (gfx1250) — compile-verified
//
#include <hip/hip_runtime.h>
#include <math.h>

// ---------------------------------------------------------------------------
// MLP_learner: h = relu(x@W0^T+b0) @ W1^T + b1; h = l2norm(h);
//              sim = h@h^T; keep top-31 per row; relu.
// N=8192, D=512, k+1=31.
//
// Exact-f32 tensor path V_WMMA_F32_16X16X4_F32 (top-31-of-8192 selection is
// too tie-sensitive for bf16). H (16MB) fits in 192MB L2 -> stream from L2,
// no LDS staging; spend registers on blocking instead:
//   wave tile 64x64 = 4x4 C tiles (128 acc VGPRs), 16 WMMAs per 8 b64 loads,
//   ping-pong double buffering over K so loads overlap the WMMA pipe.
// ---------------------------------------------------------------------------

typedef __attribute__((ext_vector_type(2))) float v2f;
typedef __attribute__((ext_vector_type(8))) float v8f;

#define FNEG_MAX (-3.402823466e+38f)

#define WMMA_F32(a, b, c)                                              \
  __builtin_amdgcn_wmma_f32_16x16x4_f32(false, (a), false, (b),        \
                                        (short)0, (c), false, false)

// C[M,N] = A[M,K] * B[N,K]^T  (+bias[col]) (relu)
// Block: 256 threads = 8 waves, 128x256 C tile. Wave grid 2(M) x 4(N);
// each wave computes 64x64 = 4x4 tiles of 16x16 via f32 WMMA.
// Requires: M%128==0, N%256==0, K%8==0.
template <bool BIAS, bool RELU>
__global__ __launch_bounds__(256) void gemm_nt_wmma(
    const float* __restrict__ A, const float* __restrict__ B,
    const float* __restrict__ bias, float* __restrict__ C,
    int M, int N, int K) {
  const int lane = threadIdx.x & 31;
  const int wave = threadIdx.x >> 5;
  const int half = lane >> 4;   // A/B: selects K pair {0,1} vs {2,3}; C: M+8
  const int l16  = lane & 15;
  const int wm = wave >> 2;     // 0..1
  const int wn = wave & 3;      // 0..3
  const int row0 = blockIdx.y * 128 + wm * 64;
  const int col0 = blockIdx.x * 256 + wn * 64;

  v8f acc[4][4];
#pragma unroll
  for (int i = 0; i < 4; ++i)
#pragma unroll
    for (int j = 0; j < 4; ++j)
      acc[i][j] = (v8f){0.f, 0.f, 0.f, 0.f, 0.f, 0.f, 0.f, 0.f};

  const int koff = 2 * half;
  const float* pa[4];
  const float* pb[4];
#pragma unroll
  for (int i = 0; i < 4; ++i)
    pa[i] = A + (long)(row0 + i * 16 + l16) * K + koff;
#pragma unroll
  for (int j = 0; j < 4; ++j)
    pb[j] = B + (long)(col0 + j * 16 + l16) * K + koff;

  v2f a0[4], b0[4], a1[4], b1[4];

  // prologue: chunk 0
#pragma unroll
  for (int i = 0; i < 4; ++i) a0[i] = *(const v2f*)(pa[i]);
#pragma unroll
  for (int j = 0; j < 4; ++j) b0[j] = *(const v2f*)(pb[j]);

  int k = 0;
#pragma unroll 1
  for (; k + 8 < K; k += 8) {
    // load chunk k+4 while computing chunk k
#pragma unroll
    for (int i = 0; i < 4; ++i) a1[i] = *(const v2f*)(pa[i] + k + 4);
#pragma unroll
    for (int j = 0; j < 4; ++j) b1[j] = *(const v2f*)(pb[j] + k + 4);
#pragma unroll
    for (int i = 0; i < 4; ++i)
#pragma unroll
      for (int j = 0; j < 4; ++j) acc[i][j] = WMMA_F32(a0[i], b0[j], acc[i][j]);
    // load chunk k+8 while computing chunk k+4
#pragma unroll
    for (int i = 0; i < 4; ++i) a0[i] = *(const v2f*)(pa[i] + k + 8);
#pragma unroll
    for (int j = 0; j < 4; ++j) b0[j] = *(const v2f*)(pb[j] + k + 8);
#pragma unroll
    for (int i = 0; i < 4; ++i)
#pragma unroll
      for (int j = 0; j < 4; ++j) acc[i][j] = WMMA_F32(a1[i], b1[j], acc[i][j]);
  }
  // tail: a0 holds chunk k; load and compute final chunk k+4
#pragma unroll
  for (int i = 0; i < 4; ++i) a1[i] = *(const v2f*)(pa[i] + k + 4);
#pragma unroll
  for (int j = 0; j < 4; ++j) b1[j] = *(const v2f*)(pb[j] + k + 4);
#pragma unroll
  for (int i = 0; i < 4; ++i)
#pragma unroll
    for (int j = 0; j < 4; ++j) acc[i][j] = WMMA_F32(a0[i], b0[j], acc[i][j]);
#pragma unroll
  for (int i = 0; i < 4; ++i)
#pragma unroll
    for (int j = 0; j < 4; ++j) acc[i][j] = WMMA_F32(a1[i], b1[j], acc[i][j]);

  // Epilogue: C/D layout: VGPR v, lanes 0-15 -> M=v, N=lane; lanes 16-31 -> M=v+8.
#pragma unroll
  for (int i = 0; i < 4; ++i) {
#pragma unroll
    for (int j = 0; j < 4; ++j) {
      const int col = col0 + j * 16 + l16;
      float bv = 0.f;
      if (BIAS) bv = bias[col];
#pragma unroll
      for (int v = 0; v < 8; ++v) {
        const int row = row0 + i * 16 + v + 8 * half;
        float val = acc[i][j][v] + bv;
        if (RELU) val = fmaxf(val, 0.f);
        C[(long)row * N + col] = val;
      }
    }
  }
}

// Row-wise L2 normalize: h /= max(||h||, 1e-12)
__global__ __launch_bounds__(256) void row_normalize(float* __restrict__ H, int D) {
  __shared__ float red[256];
  const int tid = threadIdx.x;
  float* p = H + (long)blockIdx.x * D;
  float s = 0.f;
  for (int j = tid; j < D; j += 256) {
    float v = p[j];
    s += v * v;
  }
  red[tid] = s;
  __syncthreads();
  for (int st = 128; st > 0; st >>= 1) {
    if (tid < st) red[tid] += red[tid + st];
    __syncthreads();
  }
  const float scale = 1.0f / fmaxf(sqrtf(red[0]), 1e-12f);
  for (int j = tid; j < D; j += 256) p[j] *= scale;
}

// Per-row top-k mask + relu, in place. Iterative argmax with min-index
// tie-break exactly reproduces jax.lax.top_k's duplicate handling.
__global__ __launch_bounds__(256) void topk_mask_relu(float* __restrict__ S,
                                                      int n, int k) {
  __shared__ float vals[8192];          // 32 KB (row staged in LDS)
  __shared__ unsigned char flag[8192];  // 8 KB
  __shared__ float rv[256];
  __shared__ int ri[256];
  const int tid = threadIdx.x;
  float* row = S + (long)blockIdx.x * n;

  for (int j = tid; j < n; j += 256) {
    vals[j] = row[j];
    flag[j] = 0;
  }
  __syncthreads();

  for (int it = 0; it < k; ++it) {
    float bv = FNEG_MAX;
    int bi = 0x7fffffff;
    for (int j = tid; j < n; j += 256) {
      float v = vals[j];
      if (v > bv) { bv = v; bi = j; }  // ascending j with '>' keeps min index on ties
    }
    rv[tid] = bv;
    ri[tid] = bi;
    __syncthreads();
    for (int st = 128; st > 0; st >>= 1) {
      if (tid < st) {
        float ov = rv[tid + st];
        int oi = ri[tid + st];
        if (ov > rv[tid] || (ov == rv[tid] && oi < ri[tid])) {
          rv[tid] = ov;
          ri[tid] = oi;
        }
      }
      __syncthreads();
    }
    if (tid == 0) {
      flag[ri[0]] = 1;
      vals[ri[0]] = FNEG_MAX;  // exclude from further rounds
    }
    __syncthreads();
  }

  for (int j = tid; j < n; j += 256) {
    const float g = row[j];
    row[j] = (flag[j] && g > 0.f) ? g : 0.f;
  }
}

extern "C" void kernel_launch(void* const* d_in, const int* in_sizes, int n_in,
                              void* d_out, int out_size, void* d_ws, size_t ws_size,
                              hipStream_t stream) {
  const int N = 8192, D = 512, KNB = 30;
  const float* x  = (const float*)d_in[0];
  const float* W0 = (const float*)d_in[1];
  const float* b0 = (const float*)d_in[2];
  const float* W1 = (const float*)d_in[3];
  const float* b1 = (const float*)d_in[4];
  float* out = (float*)d_out;

  float* h0 = (float*)d_ws;               // 16 MB
  float* h1 = h0 + (size_t)N * D;         // 16 MB

  dim3 blk(256);

  // h0 = relu(x @ W0^T + b0)        [8192,512] x [512,512]
  gemm_nt_wmma<true, true>
      <<<dim3(D / 256, N / 128), blk, 0, stream>>>(x, W0, b0, h0, N, D, D);

  // h1 = h0 @ W1^T + b1
  gemm_nt_wmma<true, false>
      <<<dim3(D / 256, N / 128), blk, 0, stream>>>(h0, W1, b1, h1, N, D, D);

  // h1 <- l2-normalized rows
  row_normalize<<<dim3(N), blk, 0, stream>>>(h1, D);

  // sim = h1 @ h1^T -> d_out        [8192,512] x [512,8192]  (dominant GEMM)
  gemm_nt_wmma<false, false>
      <<<dim3(N / 256, N / 128), blk, 0, stream>>>(h1, h1, nullptr, out, N, N, D);

  // top-(k+1) mask + relu, in place on d_out
  topk_mask_relu<<<dim3(N), blk, 0, stream>>>(out, N, KNB + 1);
}